// AgentAttention_83047487635717
// MI455X (gfx1250) — compile-verified
//
#include <hip/hip_runtime.h>
#include <math.h>

// ---- problem constants (match reference) ----
#define B_    512
#define NTOK  197
#define C_    384
#define H_    12
#define D_    32
#define A_    49
#define W_    14
#define MROWS (B_*NTOK)          // 100864, divisible by 64
#define SCALE 0.17677669529663687f  // D^-0.5

typedef float v8f __attribute__((ext_vector_type(8)));
typedef float v2f __attribute__((ext_vector_type(2)));

#if defined(__HIP_DEVICE_COMPILE__) && __has_builtin(__builtin_amdgcn_wmma_f32_16x16x4_f32)
#define USE_WMMA_F32 1
#endif

// =====================================================================
// GEMM: Cout[M,N] = Amat[M,K] @ Bw[N,K]^T (+ bias[N])
// M % 64 == 0, N % 64 == 0, K % 32 == 0. 256 threads = 8 waves.
// K-tile 32 -> 16 v_wmma_f32_16x16x4_f32 per barrier round.
// Both A and B staged in LDS as [row/col][k] padded to 33 so every
// fragment is an adjacent dword pair (single ds_load_2addr, no movs).
// =====================================================================
__global__ __launch_bounds__(256) void gemm_nt_kernel(
    const float* __restrict__ Amat, const float* __restrict__ Bw,
    const float* __restrict__ bias, float* __restrict__ Cout,
    int M, int K, int N)
{
  __shared__ float As[64 * 33];   // [row][k]
  __shared__ float Bs[64 * 33];   // [col][k]  (transposed vs memory op)

  const int mBase = blockIdx.y * 64;
  const int nBase = blockIdx.x * 64;
  const int t  = threadIdx.x;
  const int lr = t >> 2;          // 0..63 : row (A) / col (B)
  const int lk = (t & 3) * 8;     // 0,8,16,24 : k sub-offset (2 float4 each)

#ifdef USE_WMMA_F32
  const int lane  = t & 31;
  const int wave  = t >> 5;
  const int wm    = wave & 3;
  const int wn    = wave >> 2;
  const int lrow  = lane & 15;
  const int khalf = (lane >> 4) << 1;   // 0 or 2 (ISA A/B half-wave K split)
  v8f acc0 = {0.f,0.f,0.f,0.f,0.f,0.f,0.f,0.f};
  v8f acc1 = {0.f,0.f,0.f,0.f,0.f,0.f,0.f,0.f};
#else
  const int tx = t & 15;          // 4-col group
  const int ty = t >> 4;          // 4-row group
  float acc[4][4] = {};
#endif

  for (int k0 = 0; k0 < K; k0 += 32) {
    const float* aSrc = Amat + (size_t)(mBase + lr) * K + k0 + lk;
    const float* bSrc = Bw   + (size_t)(nBase + lr) * K + k0 + lk;
    const float4 a0 = *(const float4*)(aSrc);
    const float4 a1 = *(const float4*)(aSrc + 4);
    const float4 b0v = *(const float4*)(bSrc);
    const float4 b1v = *(const float4*)(bSrc + 4);
    float* ad = As + lr * 33 + lk;
    float* bd = Bs + lr * 33 + lk;
    ad[0] = a0.x;  ad[1] = a0.y;  ad[2] = a0.z;  ad[3] = a0.w;
    ad[4] = a1.x;  ad[5] = a1.y;  ad[6] = a1.z;  ad[7] = a1.w;
    bd[0] = b0v.x; bd[1] = b0v.y; bd[2] = b0v.z; bd[3] = b0v.w;
    bd[4] = b1v.x; bd[5] = b1v.y; bd[6] = b1v.z; bd[7] = b1v.w;
    if (k0 + 32 < K) {  // pull next tiles toward the WGP
      __builtin_prefetch(aSrc + 32, 0, 3);
      __builtin_prefetch(bSrc + 32, 0, 3);
    }
    __syncthreads();

#ifdef USE_WMMA_F32
#pragma unroll
    for (int kk = 0; kk < 8; ++kk) {
      const int kb = kk * 4 + khalf;
      v2f a, b0, b1;
      a[0]  = As[(wm * 16 + lrow) * 33 + kb];
      a[1]  = As[(wm * 16 + lrow) * 33 + kb + 1];
      b0[0] = Bs[(wn * 32 + lrow) * 33 + kb];
      b0[1] = Bs[(wn * 32 + lrow) * 33 + kb + 1];
      b1[0] = Bs[(wn * 32 + 16 + lrow) * 33 + kb];
      b1[1] = Bs[(wn * 32 + 16 + lrow) * 33 + kb + 1];
      acc0 = __builtin_amdgcn_wmma_f32_16x16x4_f32(false, a, false, b0,
                                                   (short)0, acc0, false, false);
      acc1 = __builtin_amdgcn_wmma_f32_16x16x4_f32(false, a, false, b1,
                                                   (short)0, acc1, false, false);
    }
#else
#pragma unroll
    for (int kk = 0; kk < 32; ++kk) {
      float bvv[4];
#pragma unroll
      for (int j = 0; j < 4; ++j) bvv[j] = Bs[(tx * 4 + j) * 33 + kk];
#pragma unroll
      for (int i = 0; i < 4; ++i) {
        const float avv = As[(ty * 4 + i) * 33 + kk];
#pragma unroll
        for (int j = 0; j < 4; ++j) acc[i][j] = fmaf(avv, bvv[j], acc[i][j]);
      }
    }
#endif
    __syncthreads();
  }

#ifdef USE_WMMA_F32
  const int rowsel = (lane >> 4) * 8;   // C/D: upper half-wave holds M+8
#pragma unroll
  for (int r = 0; r < 8; ++r) {
    const int gr = mBase + wm * 16 + rowsel + r;
    const int gc = nBase + wn * 32 + lrow;
    const float b0 = bias ? bias[gc]      : 0.f;
    const float b1 = bias ? bias[gc + 16] : 0.f;
    Cout[(size_t)gr * N + gc]      = acc0[r] + b0;
    Cout[(size_t)gr * N + gc + 16] = acc1[r] + b1;
  }
#else
#pragma unroll
  for (int i = 0; i < 4; ++i)
#pragma unroll
    for (int j = 0; j < 4; ++j) {
      const int gr = mBase + ty * 4 + i;
      const int gc = nBase + tx * 4 + j;
      Cout[(size_t)gr * N + gc] = acc[i][j] + (bias ? bias[gc] : 0.f);
    }
#endif
}

// =====================================================================
// Bias precompute: half-pixel bilinear 7->14 (clamp-to-edge == jax
// renormalized edge weights for this config) + broadcast adds.
// pos_bias [H][A][197], agent_bias [H][197][A]
// =====================================================================
__device__ inline float bilerp7(const float* __restrict__ s, int y, int x) {
  float sy = 0.5f * (float)y - 0.25f;
  float sx = 0.5f * (float)x - 0.25f;
  float y0f = floorf(sy), x0f = floorf(sx);
  float fy = sy - y0f,    fx = sx - x0f;
  int iy0 = (int)y0f, ix0 = (int)x0f;
  int iy1 = iy0 + 1,  ix1 = ix0 + 1;
  iy0 = iy0 < 0 ? 0 : (iy0 > 6 ? 6 : iy0);
  iy1 = iy1 < 0 ? 0 : (iy1 > 6 ? 6 : iy1);
  ix0 = ix0 < 0 ? 0 : (ix0 > 6 ? 6 : ix0);
  ix1 = ix1 < 0 ? 0 : (ix1 > 6 ? 6 : ix1);
  float v00 = s[iy0 * 7 + ix0], v01 = s[iy0 * 7 + ix1];
  float v10 = s[iy1 * 7 + ix0], v11 = s[iy1 * 7 + ix1];
  return (1.f - fy) * ((1.f - fx) * v00 + fx * v01)
       +        fy  * ((1.f - fx) * v10 + fx * v11);
}

__global__ __launch_bounds__(256) void bias_kernel(
    const float* __restrict__ an,  const float* __restrict__ na,
    const float* __restrict__ ah,  const float* __restrict__ aw,
    const float* __restrict__ ha,  const float* __restrict__ wa,
    const float* __restrict__ ac,  const float* __restrict__ ca,
    float* __restrict__ pos_bias, float* __restrict__ agent_bias)
{
  int idx = blockIdx.x * 256 + threadIdx.x;
  const int total = H_ * A_ * NTOK;
  if (idx >= total) return;
  int n = idx % NTOK;
  int a = (idx / NTOK) % A_;
  int h = idx / (NTOK * A_);
  if (n == 0) {
    pos_bias[((size_t)h * A_ + a) * NTOK + 0]   = ac[h * A_ + a];
    agent_bias[((size_t)h * NTOK + 0) * A_ + a] = ca[h * A_ + a];
  } else {
    int p = n - 1, y = p / W_, x = p % W_;
    float pb = bilerp7(an + ((size_t)h * A_ + a) * 49, y, x)
             + ah[((size_t)h * A_ + a) * W_ + y]
             + aw[((size_t)h * A_ + a) * W_ + x];
    pos_bias[((size_t)h * A_ + a) * NTOK + n] = pb;
    float ab = bilerp7(na + ((size_t)h * A_ + a) * 49, y, x)
             + ha[((size_t)h * W_ + y) * A_ + a]
             + wa[((size_t)h * W_ + x) * A_ + a];
    agent_bias[((size_t)h * NTOK + n) * A_ + a] = ab;
  }
}

// =====================================================================
// Agent pooling: 2x2 mean of q spatial tokens -> agent[b][a][c]
// =====================================================================
__global__ __launch_bounds__(256) void agent_pool_kernel(
    const float* __restrict__ qkv, float* __restrict__ agent)
{
  size_t idx = (size_t)blockIdx.x * 256 + threadIdx.x;
  if (idx >= (size_t)B_ * A_ * C_) return;
  int c = idx % C_;
  int a = (int)(idx / C_) % A_;
  int b = (int)(idx / ((size_t)C_ * A_));
  int ay = a / 7, ax = a % 7;
  float s = 0.f;
#pragma unroll
  for (int iy = 0; iy < 2; ++iy)
#pragma unroll
    for (int ix = 0; ix < 2; ++ix) {
      int n = 1 + (2 * ay + iy) * W_ + (2 * ax + ix);
      s += qkv[((size_t)(b * NTOK + n) * 3 + 0) * C_ + c];
    }
  agent[idx] = 0.25f * s;
}

__device__ inline float wave_max(float v) {
#pragma unroll
  for (int off = 16; off > 0; off >>= 1) v = fmaxf(v, __shfl_xor(v, off));
  return v;
}
__device__ inline float wave_sum(float v) {
#pragma unroll
  for (int off = 16; off > 0; off >>= 1) v += __shfl_xor(v, off);
  return v;
}

// =====================================================================
// Agent attention: per (b,h) block.
// attn = softmax(agS @ K^T + pos_bias) -> d_out agent_rep region
// agent_v = attn @ V -> ws
// =====================================================================
__global__ __launch_bounds__(256) void agent_attn_kernel(
    const float* __restrict__ agent, const float* __restrict__ qkv,
    const float* __restrict__ pos_bias, float* __restrict__ attn_out,
    float* __restrict__ agent_v)
{
  __shared__ float Ks[NTOK * 33];     // padded rows -> conflict-free per-lane rows
  __shared__ float Vs[NTOK * 32];
  __shared__ float AgS[A_ * 33];
  __shared__ float rowBuf[8 * NTOK];
  const int bh = blockIdx.x;
  const int b = bh / H_, h = bh % H_;
  const int t = threadIdx.x;

  for (int idx = t; idx < NTOK * 32; idx += 256) {
    int n = idx >> 5, d = idx & 31;
    size_t base = (size_t)(b * NTOK + n) * 3;
    Ks[n * 33 + d] = qkv[(base + 1) * C_ + h * D_ + d];
    Vs[n * 32 + d] = qkv[(base + 2) * C_ + h * D_ + d];
  }
  for (int idx = t; idx < A_ * 32; idx += 256) {
    int a = idx >> 5, d = idx & 31;
    AgS[a * 33 + d] = agent[((size_t)b * A_ + a) * C_ + h * D_ + d] * SCALE;
  }
  __syncthreads();

  const int wave = t >> 5, lane = t & 31;
  const float* pb = pos_bias + (size_t)h * A_ * NTOK;

  for (int a = wave; a < A_; a += 8) {
    float s[7];
#pragma unroll
    for (int j = 0; j < 7; ++j) {
      int n = j * 32 + lane;
      if (n < NTOK) {
        float acc = pb[a * NTOK + n];
#pragma unroll
        for (int d = 0; d < 32; ++d) acc = fmaf(AgS[a * 33 + d], Ks[n * 33 + d], acc);
        s[j] = acc;
      } else s[j] = -INFINITY;
    }
    float m = s[0];
#pragma unroll
    for (int j = 1; j < 7; ++j) m = fmaxf(m, s[j]);
    m = wave_max(m);
    float sum = 0.f;
#pragma unroll
    for (int j = 0; j < 7; ++j) { s[j] = __expf(s[j] - m); sum += s[j]; }
    sum = wave_sum(sum);
    const float inv = 1.f / sum;

    float* rb = rowBuf + wave * NTOK;
    float* ao = attn_out + ((size_t)bh * A_ + a) * NTOK;
#pragma unroll
    for (int j = 0; j < 7; ++j) {
      int n = j * 32 + lane;
      if (n < NTOK) { float p = s[j] * inv; rb[n] = p; ao[n] = p; }
    }
    // agent_v row: lane == d
    float av = 0.f;
    for (int n = 0; n < NTOK; ++n) av = fmaf(rb[n], Vs[n * 32 + lane], av);
    agent_v[((size_t)bh * A_ + a) * D_ + lane] = av;
  }
}

// =====================================================================
// Q attention: per (b,h) block. out_tmp[b][n][h*32+d]
// =====================================================================
__global__ __launch_bounds__(256) void q_attn_kernel(
    const float* __restrict__ qkv, const float* __restrict__ agent,
    const float* __restrict__ agent_bias, const float* __restrict__ agent_v,
    float* __restrict__ out_tmp)
{
  __shared__ float AgS[A_ * 33];
  __shared__ float AVs[A_ * 33];
  __shared__ float qrow[8 * 32];
  __shared__ float attnS[8 * 64];
  const int bh = blockIdx.x;
  const int b = bh / H_, h = bh % H_;
  const int t = threadIdx.x;

  for (int idx = t; idx < A_ * 32; idx += 256) {
    int a = idx >> 5, d = idx & 31;
    AgS[a * 33 + d] = agent[((size_t)b * A_ + a) * C_ + h * D_ + d];
    AVs[a * 33 + d] = agent_v[((size_t)bh * A_ + a) * D_ + d];
  }
  __syncthreads();

  const int wave = t >> 5, lane = t & 31;
  const float* abias = agent_bias + (size_t)h * NTOK * A_;
  const bool v2 = (lane + 32) < A_;
  const int a2 = v2 ? lane + 32 : A_ - 1;

  for (int n = wave; n < NTOK; n += 8) {
    qrow[wave * 32 + lane] =
        qkv[((size_t)(b * NTOK + n) * 3 + 0) * C_ + h * D_ + lane] * SCALE;
    float s1 = abias[n * A_ + lane];
    float s2 = abias[n * A_ + a2];
#pragma unroll
    for (int d = 0; d < 32; ++d) {
      const float qd = qrow[wave * 32 + d];
      s1 = fmaf(qd, AgS[lane * 33 + d], s1);
      s2 = fmaf(qd, AgS[a2   * 33 + d], s2);
    }
    float m = fmaxf(s1, v2 ? s2 : -INFINITY);
    m = wave_max(m);
    float e1 = __expf(s1 - m);
    float e2 = v2 ? __expf(s2 - m) : 0.f;
    float sum = wave_sum(e1 + e2);
    const float inv = 1.f / sum;
    attnS[wave * 64 + lane] = e1 * inv;
    if (v2) attnS[wave * 64 + lane + 32] = e2 * inv;

    // out row: lane == d
    float o = 0.f;
#pragma unroll
    for (int a = 0; a < A_; ++a) o = fmaf(attnS[wave * 64 + a], AVs[a * 33 + lane], o);
    out_tmp[(size_t)(b * NTOK + n) * C_ + h * D_ + lane] = o;
  }
}

// =====================================================================
// Depthwise 3x3 conv on v spatial tokens, accumulate into out_tmp[:,1:,:]
// =====================================================================
__global__ __launch_bounds__(256) void dwc_kernel(
    const float* __restrict__ qkv, const float* __restrict__ w_dwc,
    const float* __restrict__ b_dwc, float* __restrict__ out_tmp)
{
  size_t idx = (size_t)blockIdx.x * 256 + threadIdx.x;
  if (idx >= (size_t)B_ * 196 * C_) return;
  int c = idx % C_;
  int p = (int)(idx / C_) % 196;
  int b = (int)(idx / ((size_t)C_ * 196));
  int y = p / W_, x = p % W_;
  float acc = b_dwc[c];
#pragma unroll
  for (int dy = 0; dy < 3; ++dy) {
#pragma unroll
    for (int dx = 0; dx < 3; ++dx) {
      int yy = y + dy - 1, xx = x + dx - 1;
      if (yy >= 0 && yy < W_ && xx >= 0 && xx < W_) {
        int n = 1 + yy * W_ + xx;
        acc = fmaf(qkv[((size_t)(b * NTOK + n) * 3 + 2) * C_ + c],
                   w_dwc[c * 9 + dy * 3 + dx], acc);
      }
    }
  }
  out_tmp[((size_t)(b * NTOK) + 1 + p) * C_ + c] += acc;
}

// =====================================================================
extern "C" void kernel_launch(void* const* d_in, const int* in_sizes, int n_in,
                              void* d_out, int out_size, void* d_ws, size_t ws_size,
                              hipStream_t stream)
{
  const float* x      = (const float*)d_in[0];
  const float* w_qkv  = (const float*)d_in[1];
  const float* w_proj = (const float*)d_in[2];
  const float* b_proj = (const float*)d_in[3];
  const float* w_dwc  = (const float*)d_in[4];
  const float* b_dwc  = (const float*)d_in[5];
  const float* an     = (const float*)d_in[6];
  const float* na     = (const float*)d_in[7];
  const float* ah     = (const float*)d_in[8];
  const float* aw     = (const float*)d_in[9];
  const float* ha     = (const float*)d_in[10];
  const float* wa     = (const float*)d_in[11];
  const float* ac     = (const float*)d_in[12];
  const float* ca     = (const float*)d_in[13];
  float* out = (float*)d_out;
  float* ws  = (float*)d_ws;

  size_t off = 0;
  float* qkv        = ws + off; off += (size_t)MROWS * 3 * C_;     // 465 MB
  float* agent      = ws + off; off += (size_t)B_ * A_ * C_;       //  39 MB
  float* pos_bias   = ws + off; off += (size_t)H_ * A_ * NTOK;
  float* agent_bias = ws + off; off += (size_t)H_ * NTOK * A_;
  float* agent_v    = ws + off; off += (size_t)B_ * H_ * A_ * D_;  //  39 MB
  float* out_tmp    = ws + off; off += (size_t)MROWS * C_;         // 155 MB

  // 1) qkv = x @ w_qkv^T    (WMMA f32)
  gemm_nt_kernel<<<dim3(3 * C_ / 64, MROWS / 64), 256, 0, stream>>>(
      x, w_qkv, nullptr, qkv, MROWS, C_, 3 * C_);

  // 2) positional biases
  {
    int total = H_ * A_ * NTOK;
    bias_kernel<<<(total + 255) / 256, 256, 0, stream>>>(
        an, na, ah, aw, ha, wa, ac, ca, pos_bias, agent_bias);
  }

  // 3) agent tokens (2x2 mean pool of q)
  {
    size_t total = (size_t)B_ * A_ * C_;
    agent_pool_kernel<<<(unsigned)((total + 255) / 256), 256, 0, stream>>>(qkv, agent);
  }

  // 4) agent attention (+ agent_rep output) and agent_v
  const size_t OUT0 = (size_t)MROWS * C_;
  agent_attn_kernel<<<B_ * H_, 256, 0, stream>>>(agent, qkv, pos_bias,
                                                 out + OUT0, agent_v);

  // 5) q attention -> out_tmp
  q_attn_kernel<<<B_ * H_, 256, 0, stream>>>(qkv, agent, agent_bias, agent_v,
                                             out_tmp);

  // 6) depthwise conv accumulate
  {
    size_t total = (size_t)B_ * 196 * C_;
    dwc_kernel<<<(unsigned)((total + 255) / 256), 256, 0, stream>>>(
        qkv, w_dwc, b_dwc, out_tmp);
  }

  // 7) out = out_tmp @ w_proj^T + b_proj   (WMMA f32)
  gemm_nt_kernel<<<dim3(C_ / 64, MROWS / 64), 256, 0, stream>>>(
      out_tmp, w_proj, b_proj, out, MROWS, C_, C_);
}